// SVFCell_86517821210693
// MI455X (gfx1250) — compile-verified
//
#include <hip/hip_runtime.h>

// Elementwise SVF update, shape (16, 1048576) = 16,777,216 f32 elements.
// 7 input streams, 5 output streams -> 48 B/element, ~805 MB total.
// Pure HBM-bandwidth-bound (floor ~35 us @ 23.3 TB/s). Strategy:
//   - float4 (B128) vectorized loads/stores, fully coalesced
//   - non-temporal cache hints (streams never re-read; don't thrash L2)
//   - one float4 per thread: 4.19M threads of MLP, 7 outstanding B128
//     loads per wave before the first s_wait_loadcnt

typedef __attribute__((ext_vector_type(4))) float v4f;

__global__ __launch_bounds__(256) void svf_stream_kernel(
    const v4f* __restrict__ x,
    const v4f* __restrict__ h1,
    const v4f* __restrict__ h2,
    const v4f* __restrict__ g,
    const v4f* __restrict__ twoR,
    const v4f* __restrict__ c1,
    const v4f* __restrict__ c2,
    v4f* __restrict__ o_bp,
    v4f* __restrict__ o_lp,
    v4f* __restrict__ o_hp,
    v4f* __restrict__ o_h1,
    v4f* __restrict__ o_h2,
    int n4)
{
    const int i = blockIdx.x * blockDim.x + threadIdx.x;
    if (i >= n4) return;

    // 7 independent B128 loads issue back-to-back (non-temporal hint:
    // single-use streaming data).
    const v4f xv  = __builtin_nontemporal_load(&x[i]);
    const v4f h1v = __builtin_nontemporal_load(&h1[i]);
    const v4f h2v = __builtin_nontemporal_load(&h2[i]);
    const v4f gv  = __builtin_nontemporal_load(&g[i]);
    const v4f rv  = __builtin_nontemporal_load(&twoR[i]);
    const v4f c1v = __builtin_nontemporal_load(&c1[i]);
    const v4f c2v = __builtin_nontemporal_load(&c2[i]);

    // SVF update (per component; compiler emits v_fma_f32 chains).
    const v4f y_bp = c2v * (xv - h2v) + c1v * h1v;
    const v4f y_lp = gv * y_bp + h2v;
    const v4f y_hp = (xv - y_lp) - rv * y_bp;
    const v4f h1n  = 2.0f * y_bp - h1v;
    const v4f h2n  = 2.0f * y_lp - h2v;

    // 5 B128 non-temporal stores (write-once streams).
    __builtin_nontemporal_store(y_bp, &o_bp[i]);
    __builtin_nontemporal_store(y_lp, &o_lp[i]);
    __builtin_nontemporal_store(y_hp, &o_hp[i]);
    __builtin_nontemporal_store(h1n,  &o_h1[i]);
    __builtin_nontemporal_store(h2n,  &o_h2[i]);
}

extern "C" void kernel_launch(void* const* d_in, const int* in_sizes, int n_in,
                              void* d_out, int out_size, void* d_ws, size_t ws_size,
                              hipStream_t stream) {
    (void)n_in; (void)out_size; (void)d_ws; (void)ws_size;

    const int n  = in_sizes[0];      // 16,777,216 (divisible by 4)
    const int n4 = n / 4;

    const v4f* x    = (const v4f*)d_in[0];
    const v4f* h1   = (const v4f*)d_in[1];
    const v4f* h2   = (const v4f*)d_in[2];
    const v4f* g    = (const v4f*)d_in[3];
    const v4f* twoR = (const v4f*)d_in[4];
    const v4f* c1   = (const v4f*)d_in[5];
    const v4f* c2   = (const v4f*)d_in[6];

    // d_out: 5 outputs concatenated flat in return order.
    float* out = (float*)d_out;
    v4f* o_bp = (v4f*)(out + 0L * n);
    v4f* o_lp = (v4f*)(out + 1L * n);
    v4f* o_hp = (v4f*)(out + 2L * n);
    v4f* o_h1 = (v4f*)(out + 3L * n);
    v4f* o_h2 = (v4f*)(out + 4L * n);

    const int block = 256;                     // 8 wave32 waves per block
    const int grid  = (n4 + block - 1) / block; // 16,384 blocks

    svf_stream_kernel<<<grid, block, 0, stream>>>(
        x, h1, h2, g, twoR, c1, c2,
        o_bp, o_lp, o_hp, o_h1, o_h2, n4);
}